// TableBatchedEmbeddingBags_82557861363885
// MI455X (gfx1250) — compile-verified
//
#include <hip/hip_runtime.h>
#include <stdint.h>

// Table-batched embedding-bag forward for MI455X (gfx1250).
// One wave (32 lanes) per bag; each lane owns 2 of the 64 fp32 columns.
// Rows are staged into LDS via GLOBAL_LOAD_ASYNC_TO_LDS_B64 (one instruction ==
// one full 256B row), pipelined 24 deep per wave using ASYNCcnt.

#define WPB   8     // waves per block (256 threads)
#define SLOTS 32    // LDS row slots per wave (power of two)
#define DEPTH 24    // async pipeline depth (<= SLOTS; write slot trails read by 8)
#define D64   64    // embedding dim (reference fixes D = 64)

#define AS_GLOBAL __attribute__((address_space(1)))
#define AS_LOCAL  __attribute__((address_space(3)))

typedef int v2i __attribute__((ext_vector_type(2)));   // B64 payload type

#if defined(__HIP_DEVICE_COMPILE__) && __has_builtin(__builtin_amdgcn_global_load_async_to_lds_b64)
#define HAVE_ASYNC 1
#else
#define HAVE_ASYNC 0
#endif

#if defined(__HIP_DEVICE_COMPILE__) && __has_builtin(__builtin_amdgcn_s_wait_asynccnt)
#define WAIT_ASYNC(n) __builtin_amdgcn_s_wait_asynccnt(n)
#elif defined(__HIP_DEVICE_COMPILE__)
#define WAIT_ASYNC(n) asm volatile("s_wait_asynccnt %0" ::"i"(n) : "memory")
#else
#define WAIT_ASYNC(n)
#endif

__device__ __forceinline__ int wave_uniform(int v) {
#if defined(__HIP_DEVICE_COMPILE__)
    return __builtin_amdgcn_readfirstlane(v);
#else
    return v;
#endif
}

__global__ __launch_bounds__(WPB * 32)
void tbe_pool_fwd_kernel(const float* __restrict__ weights,
                         const int* __restrict__ table_offsets,
                         const int* __restrict__ indices,
                         const int* __restrict__ offsets,
                         float* __restrict__ out,
                         int T, int batch, int TB, int bshift) {
#if HAVE_ASYNC
    __shared__ float rowbuf[WPB][SLOTS][D64];
#endif
    const int lane = (int)threadIdx.x & 31;
    const int wave = wave_uniform((int)threadIdx.x >> 5);
    const int seg  = (int)blockIdx.x * WPB + wave;   // seg = t*batch + b
    if (seg >= TB) return;

    const int start = offsets[seg];
    const int len   = offsets[seg + 1] - start;
    // batch is a power of two in practice (bshift >= 0); division fallback kept
    // for generality (bshift < 0).
    const int t = (bshift >= 0) ? (seg >> bshift) : (seg / batch);
    const int b = seg - t * batch;
    const float* tab = weights + (long)table_offsets[t] * D64;
    const int c2 = lane * 2;

    float ax = 0.0f, ay = 0.0f;

#if HAVE_ASYNC
    if (len > 0) {
        float* buf = &rowbuf[wave][0][0];
        // Prologue: always issue exactly DEPTH row loads (tail clamps to the
        // bag's first row so the constant wait threshold below stays exact).
        for (int r = 0; r < DEPTH; ++r) {
            const int jj  = (r < len) ? (start + r) : start;
            const int idx = indices[jj];                      // wave-uniform
            const float* g = tab + (long)idx * D64 + c2;      // lane's 8B of row
            float*       l = buf + (r & (SLOTS - 1)) * D64 + c2;
            __builtin_amdgcn_global_load_async_to_lds_b64(
                (AS_GLOBAL v2i*)(uintptr_t)g,
                (AS_LOCAL  v2i*)(uint32_t)(uintptr_t)l, 0, 0);
        }
        for (int j = 0; j < len; ++j) {
            WAIT_ASYNC(DEPTH - 1);  // in-order completion => row j is in LDS
            const float2 v = *(const float2*)(buf + (j & (SLOTS - 1)) * D64 + c2);
            ax += v.x;
            ay += v.y;
            const int jn  = j + DEPTH;
            const int jj  = (jn < len) ? (start + jn) : start;  // cached dummy on tail
            const int idx = indices[jj];
            const float* g = tab + (long)idx * D64 + c2;
            float*       l = buf + (jn & (SLOTS - 1)) * D64 + c2;
            __builtin_amdgcn_global_load_async_to_lds_b64(
                (AS_GLOBAL v2i*)(uintptr_t)g,
                (AS_LOCAL  v2i*)(uint32_t)(uintptr_t)l, 0, 0);
        }
        WAIT_ASYNC(0);  // drain tail dummies before the wave exits
    }
#else
    // Fallback: direct VGPR gather with prefetch-ahead (global_prefetch_b8).
    for (int j = 0; j < len; ++j) {
        const int idx  = indices[start + j];
        const float2 v = *(const float2*)(tab + (long)idx * D64 + c2);
        ax += v.x;
        ay += v.y;
        if (j + 8 < len) {
            const int ip = indices[start + j + 8];
            __builtin_prefetch(tab + (long)ip * D64 + c2, 0, 0);
        }
    }
#endif

    // Output layout [B, T, D]: out[b][t][:]
    float2 o;
    o.x = ax;
    o.y = ay;
    *(float2*)(out + ((long)b * T + t) * D64 + c2) = o;
}

extern "C" void kernel_launch(void* const* d_in, const int* in_sizes, int n_in,
                              void* d_out, int out_size, void* d_ws, size_t ws_size,
                              hipStream_t stream) {
    (void)n_in; (void)d_ws; (void)ws_size; (void)out_size;
    const float* weights       = (const float*)d_in[0];
    const int*   table_offsets = (const int*)d_in[1];
    const int*   indices       = (const int*)d_in[2];
    const int*   offsets       = (const int*)d_in[3];
    float*       out           = (float*)d_out;

    const int T     = in_sizes[1];          // number of tables
    const int TB    = in_sizes[3] - 1;      // total segments = T * batch
    const int batch = TB / T;

    int bshift = 0;
    while ((1 << bshift) < batch && bshift < 31) ++bshift;
    if ((1 << bshift) != batch) bshift = -1;  // not a power of two -> divide

    dim3 grid((TB + WPB - 1) / WPB);
    dim3 block(WPB * 32);
    tbe_pool_fwd_kernel<<<grid, block, 0, stream>>>(
        weights, table_offsets, indices, offsets, out, T, batch, TB, bshift);
}